// ExportEquiDeformableAttn_31044023615890
// MI455X (gfx1250) — compile-verified
//
#include <hip/hip_runtime.h>
#include <cstdint>

typedef __attribute__((ext_vector_type(2))) float v2f;
typedef __attribute__((ext_vector_type(8))) float v8f;

#define CCH 32      // channels
#define SS  8       // samples per point
#define RR  128     // plane resolution
#define RR2 (RR*RR)

static __device__ __forceinline__ int imin(int a, int b) { return a < b ? a : b; }
static __device__ __forceinline__ int imax(int a, int b) { return a > b ? a : b; }

// D = A(16x4) x B(4x16) + C, f32, wave32.
static __device__ __forceinline__ v8f wmma4(v2f a, v2f b, v8f c) {
    return __builtin_amdgcn_wmma_f32_16x16x4_f32(
        /*neg_a=*/false, a, /*neg_b=*/false, b,
        /*c_mod=*/(short)0, c, /*reuse_a=*/false, /*reuse_b=*/false);
}

// Async copy of 16 bytes (per lane) from global to LDS; tracked by ASYNCcnt.
// ldsByte = byte offset within the workgroup LDS allocation (low 32 bits of a
// generic pointer to __shared__); gByteOff = per-lane byte offset vs. sgpr base.
static __device__ __forceinline__ void async_b128_to_lds(
    uint32_t ldsByte, uint32_t gByteOff, const float* __restrict__ base) {
    asm volatile("global_load_async_to_lds_b128 %0, %1, %2"
                 :: "v"(ldsByte), "v"(gByteOff), "s"(base) : "memory");
}
static __device__ __forceinline__ void wait_asynccnt0() {
    asm volatile("s_wait_asynccnt 0x0" ::: "memory");
}

// grid_sample-style bilinear, border clamp, align_corners=True. u->x(width), v->y(height).
static __device__ __forceinline__ float bilin(const float* __restrict__ pl, float u, float v) {
    float x = fminf(fmaxf(u, 0.f), 1.f) * (float)(RR - 1);
    float y = fminf(fmaxf(v, 0.f), 1.f) * (float)(RR - 1);
    float x0f = floorf(x), y0f = floorf(y);
    int x0 = imin(imax((int)x0f, 0), RR - 1);
    int y0 = imin(imax((int)y0f, 0), RR - 1);
    int x1 = imin(x0 + 1, RR - 1);
    int y1 = imin(y0 + 1, RR - 1);
    float wx = x - x0f, wy = y - y0f;
    float f00 = pl[y0 * RR + x0];
    float f01 = pl[y0 * RR + x1];
    float f10 = pl[y1 * RR + x0];
    float f11 = pl[y1 * RR + x1];
    float top = f00 * (1.f - wx) + f01 * wx;
    float bot = f10 * (1.f - wx) + f11 * wx;
    return top * (1.f - wy) + bot * wy;
}

// ---------------- Kernel A: triplane sample at query points -> feature ----------------
// One wave per point, lane = channel (coords wave-uniform; 32 parallel channel gathers).
__global__ __launch_bounds__(256) void k_sample_base(
    const float* __restrict__ qp, const float* __restrict__ cxz,
    const float* __restrict__ cxy, const float* __restrict__ cyz,
    float* __restrict__ feat, int NP, int ns)
{
    int wave = blockIdx.x * (blockDim.x >> 5) + (threadIdx.x >> 5);
    int lane = threadIdx.x & 31;
    if (wave >= NP) return;
    int b = wave / ns;
    float px = qp[(size_t)wave * 3 + 0];
    float py = qp[(size_t)wave * 3 + 1];
    float pz = qp[(size_t)wave * 3 + 2];
    size_t pb = (size_t)b * CCH * RR2 + (size_t)lane * RR2;
    float f = bilin(cxz + pb, px, pz)   // XZ plane: x=p0, y=p2
            + bilin(cxy + pb, px, py)   // XY plane: x=p0, y=p1
            + bilin(cyz + pb, py, pz);  // YZ plane: x=p1, y=p2
    feat[(size_t)wave * CCH + lane] = f;
}

// ---------------- Kernel B: feature @ [W_off | W_w] via WMMA -> pts, w ----------------
// Wcat(32x32) = [W_off(32x24) | W_w(32x8)] is packed into LDS once per block via
// async global->LDS b128 copies (the pack *is* the per-lane LDS scatter address).
__global__ __launch_bounds__(256) void k_offsets(
    const float* __restrict__ feat, const float* __restrict__ qp,
    const float* __restrict__ Woff, const float* __restrict__ boff,
    const float* __restrict__ Ww,   const float* __restrict__ bw,
    float* __restrict__ pts, float* __restrict__ wbuf, int NP)
{
    __shared__ __align__(16) float wcat[CCH * CCH];   // [c*32 + j], 4KB
    int wslot = threadIdx.x >> 5;
    int lane = threadIdx.x & 31;
    int wave = blockIdx.x * (blockDim.x >> 5) + wslot;
    int p0 = wave * 16;

    if (wslot == 0) {
        uint32_t wbase = (uint32_t)(uintptr_t)(&wcat[0]);
        // W_off: 32 rows x 6 x 16B chunks (row byte pitch 96 -> 16B aligned)
        #pragma unroll
        for (int it = 0; it < 6; ++it) {
            int idx = it * 32 + lane;          // 0..191
            int c = idx / 6, k = idx % 6;
            async_b128_to_lds(wbase + (uint32_t)(c * 128 + k * 16),
                              (uint32_t)((c * 24 + k * 4) * 4), Woff);
        }
        // W_w: 32 rows x 2 x 16B chunks -> Wcat cols 24..31
        #pragma unroll
        for (int it = 0; it < 2; ++it) {
            int idx = it * 32 + lane;          // 0..63
            int c = idx >> 1, k = idx & 1;
            async_b128_to_lds(wbase + (uint32_t)(c * 128 + 96 + k * 16),
                              (uint32_t)((c * 8 + k * 4) * 4), Ww);
        }
        wait_asynccnt0();
    }
    __syncthreads();
    if (p0 >= NP) return;                 // wave-uniform: EXEC all-1s inside
    int row = lane & 15, hi = lane >> 4;
    int j0 = row, j1 = 16 + row;

    // A fragment: lane holds feature[row][4kk + 2*hi .. +1]
    v2f a[8];
    const float* ar = feat + (size_t)imin(p0 + row, NP - 1) * CCH + 2 * hi;
    #pragma unroll
    for (int kk = 0; kk < 8; ++kk)
        a[kk] = *(const v2f*)(ar + 4 * kk);

    v8f d0 = {}, d1 = {};
    #pragma unroll
    for (int kk = 0; kk < 8; ++kk) {
        int c = 4 * kk + 2 * hi;
        v2f b0, b1;
        b0[0] = wcat[c * CCH + j0];       b0[1] = wcat[(c + 1) * CCH + j0];
        b1[0] = wcat[c * CCH + j1];       b1[1] = wcat[(c + 1) * CCH + j1];
        d0 = wmma4(a[kk], b0, d0);
        d1 = wmma4(a[kk], b1, d1);
    }

    // Scatter: cols 0..23 -> deformed points (off + query), cols 24..31 -> sample weights.
    int s0 = j0 / 3, dc0 = j0 % 3;
    #pragma unroll
    for (int i = 0; i < 8; ++i) {
        int m = i + 8 * hi;
        int p = p0 + m;
        if (p < NP) {
            pts[((size_t)p * SS + s0) * 3 + dc0] =
                d0[i] + boff[j0] + qp[(size_t)p * 3 + dc0];
            if (j1 < 24) {
                int s1 = j1 / 3, dc1 = j1 % 3;
                pts[((size_t)p * SS + s1) * 3 + dc1] =
                    d1[i] + boff[j1] + qp[(size_t)p * 3 + dc1];
            } else {
                wbuf[(size_t)p * SS + (j1 - 24)] = d1[i] + bw[j1 - 24];
            }
        }
    }
}

// ---------------- Kernel C: sample at deformed points, fused S-reduction ----------------
// r[p][c] = sum_s w[p][s] * triplane(pts[p][s])[c] ; sw[p] = sum_s w[p][s]
// (exact because W_v is linear; kills the 100MB aux buffer and 8x the GEMM rows)
__global__ __launch_bounds__(256) void k_sample_deform(
    const float* __restrict__ pts, const float* __restrict__ wbuf,
    const float* __restrict__ cxz, const float* __restrict__ cxy,
    const float* __restrict__ cyz,
    float* __restrict__ rbuf, float* __restrict__ swbuf, int NP, int ns)
{
    int wave = blockIdx.x * (blockDim.x >> 5) + (threadIdx.x >> 5);
    int lane = threadIdx.x & 31;
    if (wave >= NP) return;
    __builtin_prefetch(pts + (size_t)wave * SS * 3, 0, 0);   // global_prefetch
    __builtin_prefetch(wbuf + (size_t)wave * SS, 0, 0);
    int b = wave / ns;
    size_t pb = (size_t)b * CCH * RR2 + (size_t)lane * RR2;
    const float* pxz = cxz + pb;
    const float* pxy = cxy + pb;
    const float* pyz = cyz + pb;
    float racc = 0.f, sw = 0.f;
    #pragma unroll
    for (int s = 0; s < SS; ++s) {
        size_t q = (size_t)wave * SS + s;
        float px = pts[q * 3 + 0];
        float py = pts[q * 3 + 1];
        float pz = pts[q * 3 + 2];
        float ws = wbuf[q];
        float f = bilin(pxz, px, pz) + bilin(pxy, px, py) + bilin(pyz, py, pz);
        racc += ws * f;
        sw += ws;
    }
    rbuf[(size_t)wave * CCH + lane] = racc;
    if (lane == 0) swbuf[wave] = sw;
}

// ---------------- Kernel D: (r @ W_v + b_v*sw) @ W_out + b_out + feature ----------------
// W_v / W_out async-staged to LDS once per block; two chained WMMA GEMMs with an
// LDS (stride-33) transpose converting D-layout (lane=col) -> A-layout (lane=row).
__global__ __launch_bounds__(256) void k_head(
    const float* __restrict__ rbuf, const float* __restrict__ swbuf,
    const float* __restrict__ feat,
    const float* __restrict__ Wv,   const float* __restrict__ bv,
    const float* __restrict__ Wout, const float* __restrict__ bout,
    float* __restrict__ out, int NP)
{
    __shared__ __align__(16) float wv_s[CCH * CCH];
    __shared__ __align__(16) float wo_s[CCH * CCH];
    __shared__ __align__(16) float tile[8][16 * 33];
    int wslot = threadIdx.x >> 5;
    int wave = blockIdx.x * (blockDim.x >> 5) + wslot;
    int lane = threadIdx.x & 31;
    int p0 = wave * 16;

    if (wslot == 0) {
        uint32_t vb = (uint32_t)(uintptr_t)(&wv_s[0]);
        uint32_t ob = (uint32_t)(uintptr_t)(&wo_s[0]);
        #pragma unroll
        for (int it = 0; it < 8; ++it) {          // 1024 floats = 256 x 16B chunks
            uint32_t byt = (uint32_t)((it * 32 + lane) * 16);
            async_b128_to_lds(vb + byt, byt, Wv);
            async_b128_to_lds(ob + byt, byt, Wout);
        }
        wait_asynccnt0();
    }
    __syncthreads();
    if (p0 >= NP) return;
    int row = lane & 15, hi = lane >> 4;
    int j0 = row, j1 = 16 + row;

    v2f a[8];
    const float* ar = rbuf + (size_t)imin(p0 + row, NP - 1) * CCH + 2 * hi;
    #pragma unroll
    for (int kk = 0; kk < 8; ++kk)
        a[kk] = *(const v2f*)(ar + 4 * kk);

    v8f d0 = {}, d1 = {};
    #pragma unroll
    for (int kk = 0; kk < 8; ++kk) {
        int c = 4 * kk + 2 * hi;
        v2f b0, b1;
        b0[0] = wv_s[c * CCH + j0];     b0[1] = wv_s[(c + 1) * CCH + j0];
        b1[0] = wv_s[c * CCH + j1];     b1[1] = wv_s[(c + 1) * CCH + j1];
        d0 = wmma4(a[kk], b0, d0);
        d1 = wmma4(a[kk], b1, d1);
    }

    // tmp = d + b_v * sw, staged to per-wave LDS tile (lane=column -> lane=row transpose).
    float bv0 = bv[j0], bv1 = bv[j1];
    #pragma unroll
    for (int i = 0; i < 8; ++i) {
        int m = i + 8 * hi;
        int p = imin(p0 + m, NP - 1);
        float sw = swbuf[p];
        tile[wslot][m * 33 + j0] = d0[i] + bv0 * sw;
        tile[wslot][m * 33 + j1] = d1[i] + bv1 * sw;
    }
    __asm__ __volatile__("" ::: "memory");  // same-wave DS ops are in-order; pin compiler order

    v8f e0 = {}, e1 = {};
    #pragma unroll
    for (int kk = 0; kk < 8; ++kk) {
        int c = 4 * kk + 2 * hi;
        v2f a2, b0, b1;
        a2[0] = tile[wslot][row * 33 + c];
        a2[1] = tile[wslot][row * 33 + c + 1];
        b0[0] = wo_s[c * CCH + j0];     b0[1] = wo_s[(c + 1) * CCH + j0];
        b1[0] = wo_s[c * CCH + j1];     b1[1] = wo_s[(c + 1) * CCH + j1];
        e0 = wmma4(a2, b0, e0);
        e1 = wmma4(a2, b1, e1);
    }

    float bo0 = bout[j0], bo1 = bout[j1];
    #pragma unroll
    for (int i = 0; i < 8; ++i) {
        int m = i + 8 * hi;
        int p = p0 + m;
        if (p < NP) {
            out[(size_t)p * CCH + j0] = e0[i] + bo0 + feat[(size_t)p * CCH + j0];
            out[(size_t)p * CCH + j1] = e1[i] + bo1 + feat[(size_t)p * CCH + j1];
        }
    }
}

extern "C" void kernel_launch(void* const* d_in, const int* in_sizes, int n_in,
                              void* d_out, int out_size, void* d_ws, size_t ws_size,
                              hipStream_t stream)
{
    const float* qp   = (const float*)d_in[0];
    const float* cxz  = (const float*)d_in[1];
    const float* cxy  = (const float*)d_in[2];
    const float* cyz  = (const float*)d_in[3];
    const float* Woff = (const float*)d_in[4];
    const float* boff = (const float*)d_in[5];
    const float* Ww   = (const float*)d_in[6];
    const float* bw   = (const float*)d_in[7];
    const float* Wv   = (const float*)d_in[8];
    const float* bv   = (const float*)d_in[9];
    const float* Wout = (const float*)d_in[10];
    const float* bout = (const float*)d_in[11];
    float* out = (float*)d_out;

    int NP = out_size / CCH;                     // bs * ns = 100000
    int bs = in_sizes[1] / (CCH * RR2);          // 2
    int ns = NP / bs;

    // workspace layout (floats): feat | pts | w | r | sw  (~38.8 MB total)
    float* ws = (float*)d_ws;
    size_t o = 0;
    float* feat = ws + o;  o += (size_t)NP * CCH;
    float* pts  = ws + o;  o += (size_t)NP * SS * 3;
    float* wbuf = ws + o;  o += (size_t)NP * SS;
    float* rbuf = ws + o;  o += (size_t)NP * CCH;
    float* swb  = ws + o;  o += (size_t)NP;
    (void)ws_size; (void)n_in;

    const int wavesPerBlk = 8;                   // 256 threads = 8 wave32
    dim3 blk(256);

    int blkA = (NP + wavesPerBlk - 1) / wavesPerBlk;
    k_sample_base<<<blkA, blk, 0, stream>>>(qp, cxz, cxy, cyz, feat, NP, ns);

    int tiles = (NP + 15) / 16;
    int blkB = (tiles + wavesPerBlk - 1) / wavesPerBlk;
    k_offsets<<<blkB, blk, 0, stream>>>(feat, qp, Woff, boff, Ww, bw, pts, wbuf, NP);

    k_sample_deform<<<blkA, blk, 0, stream>>>(pts, wbuf, cxz, cxy, cyz, rbuf, swb, NP, ns);

    k_head<<<blkB, blk, 0, stream>>>(rbuf, swb, feat, Wv, bv, Wout, bout, out, NP);
}